// LineStringInstanceGenerator_38027640439075
// MI455X (gfx1250) — compile-verified
//
#include <hip/hip_runtime.h>
#include <hip/hip_bf16.h>
#include <cstdint>

// ---------------- problem constants ----------------
#define BB 4
#define HH 768
#define WW 768
#define CC 8
#define HW (HH*WW)
#define NPEAK 256
#define MAX_STEPS 64
#define LINE_LEN (MAX_STEPS+2)     // 66
#define THRESH 0.5f
#define CAND 16384                 // candidate cap per image (set >> expected threshold-bin size)

// ---------------- d_out layout (floats, reference tuple order) ----------------
#define OUT_PY   0
#define OUT_PX   (OUT_PY  + BB*NPEAK)
#define OUT_PCLS (OUT_PX  + BB*NPEAK)
#define OUT_SC   (OUT_PCLS+ BB*NPEAK)
#define OUT_PV   (OUT_SC  + BB*NPEAK)
#define OUT_LP   (OUT_PV  + BB*NPEAK)                    // (B,NPEAK,66,2)
#define OUT_LV   (OUT_LP  + BB*NPEAK*LINE_LEN*2)         // (B,NPEAK,66)
#define OUT_RP   (OUT_LV  + BB*NPEAK*LINE_LEN)
#define OUT_RV   (OUT_RP  + BB*NPEAK*LINE_LEN*2)

typedef float v2f __attribute__((ext_vector_type(2)));
typedef float v8f __attribute__((ext_vector_type(8)));

// ---------------- workspace layout (bytes) ----------------
// score_map f32[B*HW]            @ 0            (9,437,184)
// cls_map   u8 [B*HW]            @ 9437184      (2,359,296)
// hist      i32[B*1024]          @ 11796480     (16,384)
// tb        i32[B]               @ 11812864
// cnt       i32[B]               @ 11812880
// cand      u64[B*CAND]          @ 11812896     (524,288)
// pk        i32[4][B*NPEAK]      @ 12337184     (16,384)   -> total ~12.36 MB
#define WS_SCORE 0
#define WS_CLS   9437184
#define WS_HIST  11796480
#define WS_TB    11812864
#define WS_CNT   11812880
#define WS_CAND  11812896
#define WS_PK    12337184

// =====================================================================
// K0: zero histogram + candidate counters
// =====================================================================
__global__ void init_kernel(int* __restrict__ hist, int* __restrict__ cnt) {
    int i = blockIdx.x * 256 + threadIdx.x;
    if (i < BB * 1024) hist[i] = 0;
    if (i < BB) cnt[i] = 0;
}

// =====================================================================
// K1: streaming peak map. One 16x16 output tile per block (halo 18x18).
//  - async-copy halo tile of segm_logit to LDS (CDNA5 async-to-LDS path)
//  - softmax: exp on TRANS pipe, channel-sum on the matrix pipe via
//    V_WMMA_F32_16X16X4_F32 with a ones B-matrix (16 pixels / wave / iter)
//  - 3x3 per-channel window max from LDS probs; replicate-clamp at borders
//    is exactly equivalent to 'SAME' windowing for a max.
// =====================================================================
__global__ __launch_bounds__(256)
void peaks_kernel(const float* __restrict__ segm,
                  float* __restrict__ score_map,
                  unsigned char* __restrict__ cls_map)
{
    const int bx = blockIdx.x, by = blockIdx.y, b = blockIdx.z;
    const int t = threadIdx.x;

    __shared__ __align__(16) float slog[18*18*8];   // 10,368 B  halo logits
    __shared__ float sprob[18*18*8];                // 10,368 B  halo probs

    // ---- stage 1: async halo load, 16B chunks (2 per pixel), 648 chunks ----
    const float* src = segm + (size_t)b * HW * CC;
    #pragma unroll
    for (int it = 0; it < 3; ++it) {
        int ci = t + it * 256;
        if (ci < 648) {
            int pix  = ci >> 1, half = ci & 1;
            int py_  = pix / 18, px_ = pix % 18;
            int gy   = min(max(by*16 - 1 + py_, 0), HH-1);
            int gx   = min(max(bx*16 - 1 + px_, 0), WW-1);
            const float* gp = src + (size_t)(gy*WW + gx) * CC + half*4;
            unsigned lds_off = (unsigned)(uintptr_t)(&slog[ci*4]);
            asm volatile("global_load_async_to_lds_b128 %0, %1, off"
                         :: "v"(lds_off), "v"((unsigned long long)(uintptr_t)gp)
                         : "memory");
        }
    }
    asm volatile("s_wait_asynccnt 0" ::: "memory");
    __syncthreads();

    // ---- stage 2: softmax probs for all 324 halo pixels ----
    // wave handles 16 pixels/iter: lanes 0-15 own channels {0,1,4,5} of pixel
    // (lane&15), lanes 16-31 own {2,3,6,7}. This matches the f32 16x4 A-matrix
    // VGPR layout, so two chained WMMAs give D[m][*] = sum_c exp(l_c) per pixel.
    const int lane = t & 31;
    const int wv   = t >> 5;                 // wave id 0..7
    #pragma unroll
    for (int it = 0; it < 3; ++it) {
        int g = wv + it * 8;                 // wave-uniform group id
        if (g * 16 < 324) {                  // wave-uniform branch: EXEC all-ones inside
            int p  = min(g*16 + (lane & 15), 323);
            int cb = (lane < 16) ? 0 : 2;
            const float* L = &slog[p*8];
            float l0 = L[cb], l1 = L[cb+1], l4 = L[cb+4], l5 = L[cb+5];
            float mo = fmaxf(fmaxf(l0, l1), fmaxf(l4, l5));
            float mx = fmaxf(mo, __shfl_xor(mo, 16));
            float e0 = __expf(l0-mx), e1 = __expf(l1-mx);
            float e4 = __expf(l4-mx), e5 = __expf(l5-mx);
            v2f a0 = {e0, e1};
            v2f a1 = {e4, e5};
            v2f ones = {1.0f, 1.0f};         // B = 4x16 ones -> row sums
            v8f acc = {};
            acc = __builtin_amdgcn_wmma_f32_16x16x4_f32(false, a0, false, ones,
                                                        (short)0, acc, false, false);
            acc = __builtin_amdgcn_wmma_f32_16x16x4_f32(false, a1, false, ones,
                                                        (short)0, acc, false, false);
            // D layout: vgpr r, lanes 0-15 -> M=r ; lanes 16-31 -> M=8+r
            float sown = acc[0];
            #pragma unroll
            for (int i = 1; i < 8; ++i) sown = ((lane & 7) == i) ? acc[i] : sown;
            float spart = __shfl_xor(sown, 16);
            bool  own   = (lane < 8) || (lane >= 24);
            float S     = own ? sown : spart;
            float inv   = 1.0f / S;
            sprob[p*8 + cb    ] = e0 * inv;
            sprob[p*8 + cb + 1] = e1 * inv;
            sprob[p*8 + cb + 4] = e4 * inv;
            sprob[p*8 + cb + 5] = e5 * inv;
        }
    }
    __syncthreads();

    // ---- stage 3: 3x3 per-channel local max + threshold -> score/cls ----
    int ty = t >> 4, tx = t & 15;
    int hy = ty + 1, hx = tx + 1;
    float best = -1.0f; int bcls = 0;
    #pragma unroll
    for (int c = 0; c < 8; ++c) {
        float pr = sprob[(hy*18 + hx)*8 + c];
        float wmax = -1.0f;
        #pragma unroll
        for (int dy = -1; dy <= 1; ++dy)
            #pragma unroll
            for (int dx = -1; dx <= 1; ++dx)
                wmax = fmaxf(wmax, sprob[((hy+dy)*18 + (hx+dx))*8 + c]);
        bool peak = (pr == wmax) && (pr > THRESH);
        float v = peak ? pr : -1.0f;
        if (v > best) { best = v; bcls = c; }   // strict > keeps first index (argmax)
    }
    int gy = by*16 + ty, gx = bx*16 + tx;
    size_t flat = (size_t)b * HW + (size_t)(gy*WW + gx);
    score_map[flat] = best;
    cls_map[flat]   = (unsigned char)bcls;
}

// =====================================================================
// K2: 1024-bin histogram of score float-bits in (0.5, 1.0]
// =====================================================================
__device__ __forceinline__ int score_bin(float s) {
    unsigned bits = __float_as_uint(s);           // positive floats: monotonic
    return min((int)((bits - 0x3F000000u) >> 13), 1023);
}

__global__ __launch_bounds__(256)
void hist_kernel(const float* __restrict__ score_map, int* __restrict__ hist)
{
    __shared__ int h[1024];
    int t = threadIdx.x, b = blockIdx.y;
    for (int i = t; i < 1024; i += 256) h[i] = 0;
    __syncthreads();
    int base = blockIdx.x * 1024;                 // 576 blocks * 1024 = HW
    #pragma unroll
    for (int i = 0; i < 4; ++i) {
        int pix = base + i*256 + t;
        float s = score_map[(size_t)b*HW + pix];
        if (s > THRESH) atomicAdd(&h[score_bin(s)], 1);
    }
    __syncthreads();
    for (int i = t; i < 1024; i += 256)
        if (h[i]) atomicAdd(&hist[b*1024 + i], h[i]);
}

// =====================================================================
// K3: find threshold bin tb = max k with suffix-count(k) >= NPEAK
// =====================================================================
__global__ void thresh_kernel(const int* __restrict__ hist, int* __restrict__ tb)
{
    int b = blockIdx.x;
    if (threadIdx.x == 0) {
        int s = 0, r = 0;
        for (int k = 1023; k >= 0; --k) {
            s += hist[b*1024 + k];
            if (s >= NPEAK) { r = k; break; }
        }
        tb[b] = r;
    }
}

// =====================================================================
// K4: append candidates in bins >= tb (superset of the exact top-256).
// Composite key: ascending order == (score desc, flat idx asc) == lax.top_k.
// =====================================================================
__global__ __launch_bounds__(256)
void cand_kernel(const float* __restrict__ score_map,
                 const unsigned char* __restrict__ cls_map,
                 const int* __restrict__ tb,
                 unsigned long long* __restrict__ cand, int* __restrict__ cnt)
{
    int b = blockIdx.y, t = threadIdx.x;
    int base = blockIdx.x * 1024;
    int bt = tb[b];
    #pragma unroll
    for (int i = 0; i < 4; ++i) {
        int pix = base + i*256 + t;
        float s = score_map[(size_t)b*HW + pix];
        if (s > THRESH && score_bin(s) >= bt) {
            int slot = atomicAdd(&cnt[b], 1);
            if (slot < CAND) {
                unsigned long long key =
                    ((unsigned long long)(~__float_as_uint(s)) << 32)
                  | (unsigned)((pix << 3) | cls_map[(size_t)b*HW + pix]);
                cand[(size_t)b*CAND + slot] = key;
            }
        }
    }
}

// =====================================================================
// K5: deterministic exact top-256: 256 rounds of "min key > last".
// Tail slots (< 256 peaks) replicate top_k's -1 entries at smallest flat idx.
// =====================================================================
__global__ __launch_bounds__(256)
void topk_kernel(const unsigned long long* __restrict__ cand,
                 const int* __restrict__ cnt,
                 const float* __restrict__ score_map,
                 float* __restrict__ out, int* __restrict__ pk)
{
    int b = blockIdx.x, t = threadIdx.x;
    int n = min(cnt[b], CAND);
    int neff = min(n, NPEAK);
    __shared__ unsigned long long red[256];
    const unsigned long long NONE = ~0ULL;
    unsigned long long last = 0ULL;

    for (int i = 0; i < NPEAK; ++i) {
        unsigned long long local = NONE;
        for (int j = t; j < n; j += 256) {
            unsigned long long k = cand[(size_t)b*CAND + j];
            if (k > last && k < local) local = k;
        }
        red[t] = local;
        __syncthreads();
        for (int s = 128; s > 0; s >>= 1) {
            if (t < s) red[t] = (red[t] < red[t+s]) ? red[t] : red[t+s];
            __syncthreads();
        }
        unsigned long long m = red[0];
        __syncthreads();
        if (m != NONE) {
            last = m;
            if (t == 0) {
                unsigned lo = (unsigned)(m & 0xFFFFFFFFu);
                int flat = (int)(lo >> 3);
                int cls  = (int)(lo & 7u);
                float sc = __uint_as_float(~(unsigned)(m >> 32));
                int y = flat / WW, x = flat % WW;
                int o = b*NPEAK + i;
                out[OUT_PY  + o] = (float)y;
                out[OUT_PX  + o] = (float)x;
                out[OUT_PCLS+ o] = (float)cls;
                out[OUT_SC  + o] = sc;
                out[OUT_PV  + o] = 1.0f;            // sc > THRESH by construction
                pk[0*BB*NPEAK + o] = y;  pk[1*BB*NPEAK + o] = x;
                pk[2*BB*NPEAK + o] = cls; pk[3*BB*NPEAK + o] = 1;
            }
        }
    }
    if (t == 0 && neff < NPEAK) {
        int fill = neff;
        for (int idx = 0; idx < HW && fill < NPEAK; ++idx) {
            if (!(score_map[(size_t)b*HW + idx] > THRESH)) {
                int o = b*NPEAK + fill;
                int y = idx / WW, x = idx % WW;
                out[OUT_PY+o] = (float)y;  out[OUT_PX+o] = (float)x;
                out[OUT_PCLS+o] = 0.0f;    out[OUT_SC+o] = -1.0f; out[OUT_PV+o] = 0.0f;
                pk[0*BB*NPEAK+o] = y; pk[1*BB*NPEAK+o] = x;
                pk[2*BB*NPEAK+o] = 0; pk[3*BB*NPEAK+o] = 0;
                ++fill;
            }
        }
    }
}

// =====================================================================
// K6: line tracing. One lane per (batch, peak, side) -> 2048 serial chains.
// =====================================================================
__device__ __forceinline__ int clip767(float v) {
    return (int)fminf(fmaxf(floorf(v), 0.0f), 767.0f);
}

__device__ __forceinline__ void init_side(const float* __restrict__ sg,
                                          const float* __restrict__ elog,
                                          const float* __restrict__ smap,
                                          float gy, float gx, int cls,
                                          float& oy, float& ox, bool& ok)
{
    int iy = clip767(gy), ix = clip767(gx);
    size_t pi = (size_t)(iy*WW + ix);
    float sy = gy + 0.5f + smap[pi*2 + 0];
    float sx = gx + 0.5f + smap[pi*2 + 1];
    int siy = clip767(sy), six = clip767(sx);
    size_t qi = (size_t)(siy*WW + six);
    const float* L = sg + qi*CC;                 // argmax over raw logits (amap)
    int am = 0; float bv = L[0];
    #pragma unroll
    for (int c = 1; c < 8; ++c) { float v = L[c]; if (v > bv) { bv = v; am = c; } }
    ok = (am == cls) && (elog[qi] <= 0.0f);      // sigmoid(x) <= 0.5  <=>  x <= 0
    oy = sy; ox = sx;
}

__global__ __launch_bounds__(256)
void trace_kernel(const float* __restrict__ segm,
                  const float* __restrict__ elog0, const float* __restrict__ elog1,
                  const float* __restrict__ cmap,
                  const float* __restrict__ smap0, const float* __restrict__ smap1,
                  const int* __restrict__ pk, float* __restrict__ out)
{
    int tid = blockIdx.x * 256 + threadIdx.x;     // 2048 threads
    int side = tid & 1;
    int o = tid >> 1;                             // b*NPEAK + peak
    int b = o / NPEAK;

    int y   = pk[0*BB*NPEAK + o], x = pk[1*BB*NPEAK + o];
    int cls = pk[2*BB*NPEAK + o];
    bool pv = pk[3*BB*NPEAK + o] != 0;

    const float* sg = segm  + (size_t)b*HW*CC;
    const float* e0 = elog0 + (size_t)b*HW;
    const float* e1 = elog1 + (size_t)b*HW;
    const float* cm = cmap  + (size_t)b*HW*2;
    const float* s0 = smap0 + (size_t)b*HW*2;
    const float* s1 = smap1 + (size_t)b*HW*2;

    float py_ = (float)y, px_ = (float)x;
    size_t pi = (size_t)(y*WW + x);
    float sty = py_ + cm[pi*2 + 0];
    float stx = px_ + cm[pi*2 + 1];

    const float* sI = side ? s1 : s0;
    const float* eI = side ? e1 : e0;

    float p0y, p0x; bool v0;
    init_side(sg, eI, sI, py_, px_, cls, p0y, p0x, v0);
    bool alive = v0 && pv;

    size_t pbase = (size_t)(side ? OUT_RP : OUT_LP) + (size_t)o * LINE_LEN * 2;
    size_t vbase = (size_t)(side ? OUT_RV : OUT_LV) + (size_t)o * LINE_LEN;
    out[pbase + 0] = sty;  out[pbase + 1] = stx;
    out[pbase + 2] = p0y;  out[pbase + 3] = p0x;
    out[vbase + 0] = pv    ? 1.0f : 0.0f;
    out[vbase + 1] = alive ? 1.0f : 0.0f;

    float prevy = sty, prevx = stx, cury = p0y, curx = p0x;
    for (int s = 0; s < MAX_STEPS; ++s) {
        float dy = cury - prevy, dx = curx - prevx;
        float ly, lx, ry, rx; bool lv, rv;
        init_side(sg, e0, s0, cury, curx, cls, ly, lx, lv);
        init_side(sg, e1, s1, cury, curx, cls, ry, rx, rv);
        float ldot = lv ? (ly*dy + lx*dx) : -1.0f;
        float rdot = rv ? (ry*dy + rx*dx) : -1.0f;
        bool cl = lv && ((!rv && ldot > 0.0f) || (rv && ldot > rdot && ldot > 0.0f));
        bool cr = rv && ((!lv && rdot > 0.0f) || (lv && rdot > ldot && rdot > 0.0f));
        bool nv = alive && (cl || cr);
        float nxty = cl ? ly : ry;
        float nxtx = cl ? lx : rx;
        if (nv) { prevy = cury; prevx = curx; cury = nxty; curx = nxtx; }
        out[pbase + (size_t)(s+2)*2 + 0] = nv ? nxty : 0.0f;
        out[pbase + (size_t)(s+2)*2 + 1] = nv ? nxtx : 0.0f;
        out[vbase + (s+2)] = nv ? 1.0f : 0.0f;
        alive = nv;
    }
}

// =====================================================================
extern "C" void kernel_launch(void* const* d_in, const int* in_sizes, int n_in,
                              void* d_out, int out_size, void* d_ws, size_t ws_size,
                              hipStream_t stream)
{
    (void)in_sizes; (void)n_in; (void)out_size; (void)ws_size;
    const float* segm = (const float*)d_in[0];   // (B,H,W,8)
    const float* sl0  = (const float*)d_in[1];   // (B,H,W,1)
    const float* sl1  = (const float*)d_in[2];
    const float* cmp  = (const float*)d_in[3];   // (B,H,W,2)
    const float* sp0  = (const float*)d_in[4];
    const float* sp1  = (const float*)d_in[5];
    float* out = (float*)d_out;

    char* w = (char*)d_ws;                       // needs ~12.4 MB of scratch
    float*              score_map = (float*)(w + WS_SCORE);
    unsigned char*      cls_map   = (unsigned char*)(w + WS_CLS);
    int*                hist      = (int*)(w + WS_HIST);
    int*                tb        = (int*)(w + WS_TB);
    int*                cnt       = (int*)(w + WS_CNT);
    unsigned long long* cand      = (unsigned long long*)(w + WS_CAND);
    int*                pk        = (int*)(w + WS_PK);

    init_kernel  <<<16, 256, 0, stream>>>(hist, cnt);
    peaks_kernel <<<dim3(48, 48, BB), 256, 0, stream>>>(segm, score_map, cls_map);
    hist_kernel  <<<dim3(576, BB), 256, 0, stream>>>(score_map, hist);
    thresh_kernel<<<BB, 32, 0, stream>>>(hist, tb);
    cand_kernel  <<<dim3(576, BB), 256, 0, stream>>>(score_map, cls_map, tb, cand, cnt);
    topk_kernel  <<<BB, 256, 0, stream>>>(cand, cnt, score_map, out, pk);
    trace_kernel <<<8, 256, 0, stream>>>(segm, sl0, sl1, cmp, sp0, sp1, pk, out);
}